// PseudoGrid_23527830847990
// MI455X (gfx1250) — compile-verified
//
#include <hip/hip_runtime.h>
#include <math.h>

#define NB   4
#define N1   2048
#define N2   8192
#define NS   32
#define NKP  15
#define CIN  64
#define COUT 128

#define RADIUS   0.1f
#define EXTENT   0.04f
#define BN_EPS   1e-5f

typedef __attribute__((ext_vector_type(2))) float v2f;
typedef __attribute__((ext_vector_type(8))) float v8f;
typedef __attribute__((__vector_size__(16))) int i32x4;

#define __AS1 __attribute__((address_space(1)))
#define __AS3 __attribute__((address_space(3)))

#if __has_builtin(__builtin_amdgcn_global_load_async_to_lds_b128)
#define HAVE_ASYNC_LDS 1
#else
#define HAVE_ASYNC_LDS 0
#endif

static __device__ __forceinline__ unsigned ballot32(bool p) {
  return __builtin_amdgcn_ballot_w32(p);
}

#if HAVE_ASYNC_LDS
// Builtin signature (from hipcc diagnostic): param0 = int4 addrspace(1)*,
// param1 = int4 addrspace(3)*, then imm offset, imm cpol.
// Flat LDS pointers carry the LDS byte offset in addr[31:0] (ISA 10.2), so
// truncating to a 32-bit AS3 pointer is exact; integer round-trips also
// strip const.
static __device__ __forceinline__ void async_copy16(const void* g, void* l) {
  __builtin_amdgcn_global_load_async_to_lds_b128(
      (__AS1 i32x4*)(unsigned long long)g,
      (__AS3 i32x4*)(unsigned int)(unsigned long long)l,
      0, 0);
}
static __device__ __forceinline__ void wait_async0() {
#if __has_builtin(__builtin_amdgcn_s_wait_asynccnt)
  __builtin_amdgcn_s_wait_asynccnt(0);
#else
  asm volatile("s_wait_asynccnt 0x0" ::: "memory");
#endif
}
#endif

// ---------------------------------------------------------------------------
// Kernel 0: transpose support_features [B, C, N2] -> fT [B, N2, C]
// 64x64 tile; load phase uses async global->LDS (ASYNCcnt path), store
// phase is coalesced on the channel axis.
// ---------------------------------------------------------------------------
__global__ void k_transpose(const float* __restrict__ sf, float* __restrict__ fT) {
  __shared__ __align__(16) float tile[CIN][68];   // 272B row stride: 16B-aligned
  const int b  = blockIdx.x >> 7;          // 128 j-tiles per batch
  const int j0 = (blockIdx.x & 127) * 64;
  const int t  = threadIdx.x;
#if HAVE_ASYNC_LDS
#pragma unroll
  for (int r = 0; r < 4; ++r) {
    int q  = r * 256 + t;                  // 1024 x 16B chunks
    int c  = q >> 4;
    int j4 = (q & 15) << 2;
    const float* gp = sf + (((size_t)(b * CIN + c)) << 13) + j0 + j4;
    async_copy16(gp, &tile[c][j4]);
  }
  wait_async0();
#else
#pragma unroll
  for (int r = 0; r < 16; ++r) {
    int idx = r * 256 + t;
    int c = idx >> 6, jj = idx & 63;
    tile[c][jj] = sf[((size_t)(b * CIN + c) << 13) + j0 + jj];
  }
#endif
  __syncthreads();
#pragma unroll
  for (int r = 0; r < 16; ++r) {
    int idx = r * 256 + t;
    int jj = idx >> 6, c = idx & 63;
    fT[((size_t)((b << 13) + j0 + jj)) * CIN + c] = tile[c][jj];
  }
}

// ---------------------------------------------------------------------------
// Kernel 1: per query point — ordered ball query (wave32 ballot + prefix
// popcount), kernel-point influence weights, and the per-point contraction
//   mid[m,c] = sum_s f[idx[s],c] * (sum_k w[k,s] * KW[k,c])
// Block = 64 threads (2 waves): wave 0 builds idx/w in LDS, all 64 threads
// then each own one input channel.
// ---------------------------------------------------------------------------
__global__ void k_kpconv(const float* __restrict__ qxyz,
                         const float* __restrict__ sxyz,
                         const float* __restrict__ qmask,
                         const float* __restrict__ smask,
                         const float* __restrict__ fT,
                         const float* __restrict__ kpts,
                         const float* __restrict__ kw,
                         float* __restrict__ mid) {
  const int m = blockIdx.x;                // b*N1 + n
  const int b = m >> 11;
  const int t = threadIdx.x;

  __shared__ int   s_idx[NS];
  __shared__ float s_w[NKP][NS];
  __shared__ int   s_cnt;

  const float qx = qxyz[m * 3 + 0];
  const float qy = qxyz[m * 3 + 1];
  const float qz = qxyz[m * 3 + 2];

  if (t < 32) {
    s_idx[t] = 0;
    int found = 0;
    const float r2 = RADIUS * RADIUS;
    for (int j0 = 0; j0 < N2 && found < NS; j0 += 32) {
      const int j = j0 + t;
      const float* sp = sxyz + (size_t)(b * N2 + j) * 3;
      const float dx = sp[0] - qx, dy = sp[1] - qy, dz = sp[2] - qz;
      const float d2 = dx * dx + dy * dy + dz * dz;
      const bool valid = (d2 < r2) && (smask[b * N2 + j] > 0.0f);
      const unsigned msk = ballot32(valid);
      if (valid) {
        const int slot = found + __popc(msk & ((1u << t) - 1u));
        if (slot < NS) s_idx[slot] = j;
      }
      found += __popc(msk);
    }
    if (t == 0) s_cnt = (found < NS) ? found : NS;
  }
  __syncthreads();

  if (t < 32) {
    const int cnt = s_cnt;
    const int jj  = (t < cnt) ? s_idx[t] : s_idx[0];   // pad with first nbr
    s_idx[t] = jj;
    const float qm = qmask[m];
    const float fm = ((t < cnt) ? 1.0f : 0.0f) + (1.0f - qm);
    const float* sp = sxyz + (size_t)(b * N2 + jj) * 3;
    const float rx = sp[0] - qx, ry = sp[1] - qy, rz = sp[2] - qz;
#pragma unroll
    for (int k = 0; k < NKP; ++k) {
      const float dx = rx - kpts[k * 3 + 0];
      const float dy = ry - kpts[k * 3 + 1];
      const float dz = rz - kpts[k * 3 + 2];
      const float sd = dx * dx + dy * dy + dz * dz;
      float w = 1.0f - sqrtf(sd) * (1.0f / EXTENT);
      w = fmaxf(w, 0.0f) * fm;
      s_w[k][t] = w;
    }
  }
  __syncthreads();

  // Phase B: thread t == input channel c
  const int c = t;
  float kwr[NKP];
#pragma unroll
  for (int k = 0; k < NKP; ++k) kwr[k] = kw[k * CIN + c];

  float acc = 0.0f;
  for (int s = 0; s < NS; ++s) {
    const float f = fT[(size_t)(b * N2 + s_idx[s]) * CIN + c];
    float g = 0.0f;
#pragma unroll
    for (int k = 0; k < NKP; ++k) g += s_w[k][s] * kwr[k];
    acc += g * f;
  }
  mid[(size_t)m * CIN + c] = acc;
}

// ---------------------------------------------------------------------------
// Kernel 2: Y[o, m] = sum_c W_out[o,c] * mid[m,c]   (M=8192, N=128, K=64)
// fp32 WMMA 16x16x4.  256 threads = 8 waves; wave w owns the 16-channel
// slice n0 = 16*w, block owns the 16-row slice m0 = 16*blockIdx.x.
// A tile staged via async global->LDS.
// Fragment layouts per ISA 7.12.2:
//   A 16x4 fp32 : lane L -> M = L%16 ; VGPR v -> K = v + (L/16)*2
//   B  4x16 fp32: lane L -> N = L%16 ; VGPR v -> K = v + (L/16)*2
//   C/D 16x16   : lane L -> N = L%16 ; VGPR r -> M = r + (L/16)*8
// ---------------------------------------------------------------------------
__global__ void k_gemm_wmma(const float* __restrict__ mid,
                            const float* __restrict__ Wout,
                            float* __restrict__ Y) {
  __shared__ __align__(16) float sA[16][68];   // 272B stride: 16B-aligned rows
  const int m0 = blockIdx.x * 16;
  const int t  = threadIdx.x;
#if HAVE_ASYNC_LDS
  {
    int row = t >> 4;                      // 256 x 16B chunks = 16x64 floats
    int k4  = (t & 15) << 2;
    const float* gp = mid + (size_t)(m0 + row) * CIN + k4;
    async_copy16(gp, &sA[row][k4]);
  }
  wait_async0();
#else
#pragma unroll
  for (int r = 0; r < 4; ++r) {
    int idx = r * 256 + t;
    int row = idx >> 6, k = idx & 63;
    sA[row][k] = mid[(size_t)(m0 + row) * CIN + k];
  }
#endif
  __syncthreads();

  const int wave = t >> 5;
  const int lane = t & 31;
  const int n0   = wave * 16;
  const int lm   = lane & 15;              // M (for A) / N (for B,D)
  const int lh   = lane >> 4;              // K-half selector

  v8f acc = {};
#pragma unroll
  for (int k0 = 0; k0 < CIN; k0 += 4) {
    v2f a, bb;
    a.x = sA[lm][k0 + lh * 2 + 0];
    a.y = sA[lm][k0 + lh * 2 + 1];
    const float* wp = Wout + (size_t)(n0 + lm) * CIN + k0 + lh * 2;
    bb.x = wp[0];
    bb.y = wp[1];
    acc = __builtin_amdgcn_wmma_f32_16x16x4_f32(
        /*neg_a=*/false, a, /*neg_b=*/false, bb,
        /*c_mod=*/(short)0, acc, /*reuse_a=*/false, /*reuse_b=*/false);
  }

  float* yp = Y + (size_t)(n0 + lm) * (NB * N1) + m0 + lh * 8;
#pragma unroll
  for (int r = 0; r < 8; ++r) yp[r] = acc[r];
}

// ---------------------------------------------------------------------------
// Kernel 3: per-channel sum / sum-of-squares over m (one block per channel)
// ---------------------------------------------------------------------------
__global__ void k_bn_reduce(const float* __restrict__ Y, float* __restrict__ sums) {
  const int o = blockIdx.x;
  const int t = threadIdx.x;
  const float* yp = Y + (size_t)o * (NB * N1);
  float s = 0.0f, s2 = 0.0f;
  for (int m = t; m < NB * N1; m += 256) {
    const float v = yp[m];
    s += v; s2 += v * v;
  }
  __shared__ float r1[256], r2[256];
  r1[t] = s; r2[t] = s2;
  __syncthreads();
  for (int off = 128; off > 0; off >>= 1) {
    if (t < off) { r1[t] += r1[t + off]; r2[t] += r2[t + off]; }
    __syncthreads();
  }
  if (t == 0) { sums[o] = r1[0]; sums[COUT + o] = r2[0]; }
}

// ---------------------------------------------------------------------------
// Kernel 4: BN (batch stats) + ReLU, write out [B, C_out, N1]
// ---------------------------------------------------------------------------
__global__ void k_bn_relu(const float* __restrict__ Y,
                          const float* __restrict__ sums,
                          const float* __restrict__ gamma,
                          const float* __restrict__ beta,
                          float* __restrict__ out) {
  const int bo = blockIdx.x;               // b*COUT + o
  const int b = bo >> 7, o = bo & 127;
  const float inv_n = 1.0f / (float)(NB * N1);
  const float mean  = sums[o] * inv_n;
  const float var   = sums[COUT + o] * inv_n - mean * mean;
  const float inv   = rsqrtf(var + BN_EPS);
  const float ga = gamma[o], be = beta[o];
  const float* yp = Y + (size_t)o * (NB * N1) + b * N1;
  float* op = out + (size_t)bo * N1;
  for (int n = threadIdx.x; n < N1; n += 256) {
    const float v = (yp[n] - mean) * inv * ga + be;
    op[n] = fmaxf(v, 0.0f);
  }
}

// ---------------------------------------------------------------------------
extern "C" void kernel_launch(void* const* d_in, const int* in_sizes, int n_in,
                              void* d_out, int out_size, void* d_ws, size_t ws_size,
                              hipStream_t stream) {
  const float* qxyz  = (const float*)d_in[0];
  const float* sxyz  = (const float*)d_in[1];
  const float* qm    = (const float*)d_in[2];
  const float* sm    = (const float*)d_in[3];
  const float* sf    = (const float*)d_in[4];
  const float* kpts  = (const float*)d_in[5];
  const float* kw    = (const float*)d_in[6];
  const float* Wout  = (const float*)d_in[7];
  const float* gamma = (const float*)d_in[8];
  const float* beta  = (const float*)d_in[9];
  float* out = (float*)d_out;

  float* ws   = (float*)d_ws;
  float* fT   = ws;                                   // B*N2*CIN   = 2M floats
  float* mid  = fT  + (size_t)NB * N2 * CIN;          // B*N1*CIN   = 512K floats
  float* Y    = mid + (size_t)NB * N1 * CIN;          // COUT*B*N1  = 1M floats
  float* sums = Y   + (size_t)COUT * NB * N1;         // 2*COUT floats

  k_transpose<<<NB * (N2 / 64), 256, 0, stream>>>(sf, fT);
  k_kpconv   <<<NB * N1, 64, 0, stream>>>(qxyz, sxyz, qm, sm, fT, kpts, kw, mid);
  k_gemm_wmma<<<(NB * N1) / 16, 256, 0, stream>>>(mid, Wout, Y);
  k_bn_reduce<<<COUT, 256, 0, stream>>>(Y, sums);
  k_bn_relu  <<<NB * COUT, 256, 0, stream>>>(Y, sums, gamma, beta, out);
}